// HyConv_woparam_24292335026751
// MI455X (gfx1250) — compile-verified
//
#include <hip/hip_runtime.h>

// Hypergraph conv (HyConv, no params):
//   ef[n,e,c] = (1/de[n,e]) * sum_v H[n,v,e] * x[n,v,c]   ; de = col-sums of H
//   y [n,v,c] = (1/dv[n,v]) * sum_e H[n,v,e] * ef[n,e,c]  ; dv = row-sums of H
// Memory-bound: H (512 MB) streamed twice from HBM -> ~46 us floor @ 23.3 TB/s.
// f32 WMMA 16x16x4 provides ample compute; degree sums fused into A-fragment loads.

typedef __attribute__((ext_vector_type(2))) float v2f;
typedef __attribute__((ext_vector_type(8))) float v8f;

#define N_B   2
#define V_DIM 8192
#define E_DIM 8192
#define C_DIM 16
#define WPB   4   // waves per block

// ---------------------------------------------------------------------------
// Kernel 1: vertex -> edge.  One wave owns a 16-wide e-tile; K loops over v.
// A = H^T tile (M=e, K=v): lane m reads H[(v+k)*E + e0+m]  (64B coalesced in e)
// B = x tile   (K=v, N=c): lane reads x[(v+k)*16 + c]
// de partial = sum of A fragments per lane; halves combined via shfl_xor(16).
// ---------------------------------------------------------------------------
__global__ __launch_bounds__(WPB * 32)
void k_vertex_to_edge(const float* __restrict__ x,
                      const float* __restrict__ H,
                      float* __restrict__ ef)
{
    const int tilesPerBatch = E_DIM / (16 * WPB);          // 128
    const int n    = blockIdx.x / tilesPerBatch;
    const int eb   = blockIdx.x % tilesPerBatch;
    const int wave = threadIdx.x >> 5;
    const int lane = threadIdx.x & 31;
    const int half = lane >> 4;                            // 0: K={0,1}, 1: K={2,3}
    const int m    = lane & 15;
    const int e0   = (eb * WPB + wave) * 16;

    const float* __restrict__ Hn  = H  + (size_t)n * V_DIM * E_DIM;
    const float* __restrict__ xn  = x  + (size_t)n * V_DIM * C_DIM;
    float*       __restrict__ efn = ef + (size_t)n * E_DIM * C_DIM;

    v8f   acc   = {};
    float deSum = 0.0f;

    const int aoff = (2 * half) * E_DIM + e0 + m;          // H[(v+2h)*E + e0+m]
    const int boff = (2 * half) * C_DIM + m;               // x[(v+2h)*16 + c]

    #pragma unroll 8
    for (int v = 0; v < V_DIM; v += 4) {
        v2f a, b;
        a.x = Hn[(size_t)v * E_DIM + aoff];                // K = 2h
        a.y = Hn[(size_t)v * E_DIM + aoff + E_DIM];        // K = 2h+1
        b.x = xn[v * C_DIM + boff];
        b.y = xn[v * C_DIM + boff + C_DIM];
        deSum += a.x + a.y;                                // fused column-sum of H
        acc = __builtin_amdgcn_wmma_f32_16x16x4_f32(
                  false, a, false, b, (short)0, acc, false, false);
    }

    // lane m and lane m+16 each hold half of de[e0+m]
    const float deTot = deSum + __shfl_xor(deSum, 16, 32);

    #pragma unroll
    for (int j = 0; j < 8; ++j) {
        const int   row = j + 8 * half;                    // D VGPR j: M = j (+8 for hi half)
        const float inv = 1.0f / __shfl(deTot, row, 32);   // de[e0+row] lives on lane `row`
        efn[(e0 + row) * C_DIM + m] = acc[j] * inv;
    }
}

// ---------------------------------------------------------------------------
// Kernel 2: edge -> vertex.  One wave owns a 16-wide v-tile; K loops over e.
// A = H tile (M=v, K=e): lane m reads H[(v0+m)*E + e + 2h .. +1]  -> b64 load,
//   each 128B line fully consumed over 8 consecutive K-steps (L0 resident).
// B = ef tile (K=e, N=c): rows are contiguous 64B in the 1 MB L2-resident ef.
// dv partial = sum of A fragments per lane; halves combined via shfl_xor(16).
// ---------------------------------------------------------------------------
__global__ __launch_bounds__(WPB * 32)
void k_edge_to_vertex(const float* __restrict__ ef,
                      const float* __restrict__ H,
                      float* __restrict__ y)
{
    const int tilesPerBatch = V_DIM / (16 * WPB);          // 128
    const int n    = blockIdx.x / tilesPerBatch;
    const int vb   = blockIdx.x % tilesPerBatch;
    const int wave = threadIdx.x >> 5;
    const int lane = threadIdx.x & 31;
    const int half = lane >> 4;
    const int m    = lane & 15;
    const int v0   = (vb * WPB + wave) * 16;

    const float* __restrict__ Hn  = H  + (size_t)n * V_DIM * E_DIM;
    const float* __restrict__ efn = ef + (size_t)n * E_DIM * C_DIM;
    float*       __restrict__ yn  = y  + (size_t)n * V_DIM * C_DIM;

    v8f   acc   = {};
    float dvSum = 0.0f;

    const float* __restrict__ arow = Hn + (size_t)(v0 + m) * E_DIM + 2 * half;
    const int boff = (2 * half) * C_DIM + m;

    #pragma unroll 8
    for (int e = 0; e < E_DIM; e += 4) {
        v2f a = *(const v2f*)(arow + e);                   // H[v0+m][e+2h], [e+2h+1]
        v2f b;
        b.x = efn[e * C_DIM + boff];
        b.y = efn[e * C_DIM + boff + C_DIM];
        dvSum += a.x + a.y;                                // fused row-sum of H
        acc = __builtin_amdgcn_wmma_f32_16x16x4_f32(
                  false, a, false, b, (short)0, acc, false, false);
    }

    const float dvTot = dvSum + __shfl_xor(dvSum, 16, 32);

    #pragma unroll
    for (int j = 0; j < 8; ++j) {
        const int   row = j + 8 * half;
        const float inv = 1.0f / __shfl(dvTot, row, 32);
        yn[(v0 + row) * C_DIM + m] = acc[j] * inv;
    }
}

// ---------------------------------------------------------------------------
extern "C" void kernel_launch(void* const* d_in, const int* in_sizes, int n_in,
                              void* d_out, int out_size, void* d_ws, size_t ws_size,
                              hipStream_t stream) {
    (void)in_sizes; (void)n_in; (void)out_size; (void)ws_size;
    const float* x = (const float*)d_in[0];   // [N, V, C] f32
    const float* H = (const float*)d_in[1];   // [N, V, E] f32
    float* y  = (float*)d_out;                // [N, V, C] f32
    float* ef = (float*)d_ws;                 // [N, E, C] f32 scratch (1 MB)

    const int blocks1 = N_B * (E_DIM / (16 * WPB));        // 256
    const int blocks2 = N_B * (V_DIM / (16 * WPB));        // 256

    k_vertex_to_edge<<<dim3(blocks1), dim3(WPB * 32), 0, stream>>>(x, H, ef);
    k_edge_to_vertex<<<dim3(blocks2), dim3(WPB * 32), 0, stream>>>(ef, H, y);
}